// P2P_88399016886558
// MI455X (gfx1250) — compile-verified
//
#include <hip/hip_runtime.h>
#include <hip/hip_bf16.h>

#define BB    16
#define CC    3
#define HH    224
#define WWID  224
#define HWP   (HH*WWID)       // 50176
#define EMB   16
#define NSEG  196
#define MC    16
#define NPAD  208             // 13*16 (M padding for 16x16 tiles)
#define KG    16              // gemb K padded 15 -> 16
#define JITTER 0.001f

typedef __attribute__((ext_vector_type(2))) float v2f;
typedef __attribute__((ext_vector_type(8))) float v8f;

__device__ __forceinline__ uint32_t hashu(uint32_t x) {
    x ^= x >> 17; x *= 0xed5ad4bbu;
    x ^= x >> 11; x *= 0xac4c1b51u;
    x ^= x >> 15; x *= 0x31848babu;
    x ^= x >> 14;
    return x;
}
__device__ __forceinline__ float u01(uint32_t h) {
    return ((float)h + 0.5f) * 2.3283064365386963e-10f;  // (0,1)
}
__device__ __forceinline__ float sigmoidf(float z) {
    return 1.f / (1.f + __expf(-z));
}

// ---------------- K0: zero segment accumulators ----------------
__global__ void k_zero(float* __restrict__ p, int n) {
    int i = blockIdx.x*blockDim.x + threadIdx.x;
    if (i < n) p[i] = 0.f;
}

// ---------------- K1: 1x1 conv + pixel_probs + hierarchical segment sums ----
__global__ void k_conv(const float* __restrict__ x, const int* __restrict__ groups,
                       const float* __restrict__ Wc, const float* __restrict__ bc,
                       float* __restrict__ pixel_out,
                       float* __restrict__ sums, float* __restrict__ cnts) {
    __shared__ float ls[NSEG*EMB];
    __shared__ float lc[NSEG];
    __shared__ float w[EMB*CC];
    __shared__ float bsh[EMB];
    const int b = blockIdx.y;
    const int tid = threadIdx.x;
    for (int i = tid; i < NSEG*EMB; i += blockDim.x) ls[i] = 0.f;
    for (int i = tid; i < NSEG; i += blockDim.x) lc[i] = 0.f;
    if (tid < EMB*CC) w[tid] = Wc[tid];
    if (tid < EMB)    bsh[tid] = bc[tid];
    __syncthreads();

    const int p0 = blockIdx.x * 1024;          // 49 blocks * 1024 px = 50176
    #pragma unroll
    for (int it = 0; it < 4; ++it) {
        int p = p0 + it*256 + tid;
        float x0 = x[((size_t)b*CC + 0)*HWP + p];
        float x1 = x[((size_t)b*CC + 1)*HWP + p];
        float x2 = x[((size_t)b*CC + 2)*HWP + p];
        int   g  = groups[(size_t)b*HWP + p];
        float e0 = 0.f;
        #pragma unroll
        for (int e = 0; e < EMB; ++e) {
            float v = bsh[e] + w[e*3+0]*x0 + w[e*3+1]*x1 + w[e*3+2]*x2;
            if (e == 0) e0 = v;
            atomicAdd(&ls[g*EMB + e], v);
        }
        atomicAdd(&lc[g], 1.f);
        pixel_out[(size_t)b*HWP + p] = sigmoidf(e0);
    }
    __syncthreads();
    for (int i = tid; i < NSEG*EMB; i += blockDim.x)
        atomicAdd(&sums[(size_t)b*NSEG*EMB + i], ls[i]);
    for (int i = tid; i < NSEG; i += blockDim.x)
        atomicAdd(&cnts[(size_t)b*NSEG + i], lc[i]);
}

// ---------------- K2: averages -> mu, group_probs, padded f32 gemb ---------
// gembp: [B][NPAD][KG] f32; cols 0..14 = avg[:, 1:16], col 15 = 0; pad rows 0
__global__ void k_avg(const float* __restrict__ sums, const float* __restrict__ cnts,
                      float* __restrict__ mu, float* __restrict__ gp_out,
                      float* __restrict__ gembp) {
    int idx = blockIdx.x*blockDim.x + threadIdx.x;   // B*NPAD rows
    if (idx >= BB*NPAD) return;
    int s = idx % NPAD, b = idx / NPAD;
    float* row = gembp + (size_t)idx * KG;
    if (s < NSEG) {
        float inv = 1.f / fmaxf(cnts[b*NSEG + s], 1.f);
        float m   = sums[((size_t)b*NSEG + s)*EMB + 0] * inv;
        mu[b*NSEG + s]     = m;
        gp_out[b*NSEG + s] = sigmoidf(m);
        #pragma unroll
        for (int e = 1; e < EMB; ++e)
            row[e-1] = sums[((size_t)b*NSEG + s)*EMB + e] * inv;
        row[15] = 0.f;
    } else {
        #pragma unroll
        for (int e = 0; e < KG; ++e) row[e] = 0.f;
    }
}

// ---------------- K3: sigma = G G^T + jitter I (fp32 WMMA 16x16x4) ---------
__global__ void k_sigma(const float* __restrict__ gembp,
                        float* __restrict__ sigma_out, float* __restrict__ sigma_ws) {
    int wave = (blockIdx.x*blockDim.x + threadIdx.x) >> 5;  // 16*169 waves exactly
    int lane = threadIdx.x & 31;
    int b  = wave / 169;
    int t  = wave % 169;
    int tm = t / 13, tn = t % 13;
    int hi = lane >> 4, l15 = lane & 15;
    const float* gb = gembp + (size_t)b * NPAD * KG;
    const float* rA = gb + (tm*16 + l15)*KG;   // A: 16x4 tile rows (M = lane&15)
    const float* rB = gb + (tn*16 + l15)*KG;   // B = G^T: B[k][n] = G[n][k]
    v8f c = {};
    #pragma unroll
    for (int ks = 0; ks < KG/4; ++ks) {
        int k = ks*4 + 2*hi;                   // VGPR v holds K = v + 2*hi
        v2f a, bm;
        a.x  = rA[k]; a.y  = rA[k+1];
        bm.x = rB[k]; bm.y = rB[k+1];
        c = __builtin_amdgcn_wmma_f32_16x16x4_f32(false, a, false, bm,
                                                  (short)0, c, false, false);
    }
    bool ncol = (tn*16 + l15) < NSEG;
    #pragma unroll
    for (int r = 0; r < 8; ++r) {
        int m = tm*16 + r + 8*hi;
        int n = tn*16 + l15;
        if (m < NSEG && ncol) {
            float vo = c[r] + ((m == n) ? JITTER : 0.f);
            sigma_out[((size_t)b*NSEG + m)*NSEG + n] = vo;
            sigma_ws [((size_t)b*NSEG + m)*NSEG + n] = vo;
        }
    }
}

// ---------------- K4: per-batch right-looking Cholesky; emit padded f32 L --
// Lp: [B][NPAD][NSEG], zero upper triangle + zero pad rows
__global__ void k_chol(float* __restrict__ S_all, float* __restrict__ Lp_all) {
    const int b = blockIdx.x;
    float* S = S_all + (size_t)b*NSEG*NSEG;
    __shared__ float sd;
    const int tid = threadIdx.x, nt = blockDim.x;
    for (int j = 0; j < NSEG; ++j) {
        if (tid == 0) sd = sqrtf(fmaxf(S[j*NSEG + j], 1e-30f));
        __syncthreads();
        float d = sd;
        if (tid == 0) S[j*NSEG + j] = d;
        for (int i = j+1+tid; i < NSEG; i += nt) S[i*NSEG + j] /= d;
        __syncthreads();
        for (int i = j+1+tid; i < NSEG; i += nt) {
            float lij = S[i*NSEG + j];
            for (int k = j+1; k <= i; ++k) S[i*NSEG + k] -= lij * S[k*NSEG + j];
        }
        __syncthreads();
    }
    float* Lp = Lp_all + (size_t)b * NPAD * NSEG;
    for (int idx = tid; idx < NPAD*NSEG; idx += nt) {
        int r = idx / NSEG, k = idx % NSEG;
        Lp[idx] = (r < NSEG && k <= r) ? S[r*NSEG + k] : 0.f;
    }
}

// ---------------- K5: deterministic Gaussian eps (f32, [B][NSEG][MC]) ------
__global__ void k_eps(float* __restrict__ epsp) {
    int idx = blockIdx.x*blockDim.x + threadIdx.x;   // B*NSEG*MC
    if (idx >= BB*NSEG*MC) return;
    float uu1 = u01(hashu(2u*idx + 1u));
    float uu2 = u01(hashu(2u*idx + 0x9e3779b9u));
    epsp[idx] = sqrtf(-2.f*logf(uu1)) * __cosf(6.28318530718f*uu2);
}

// ---------------- K6: logits = mu + L@eps (fp32 WMMA, K=196=49x4) ----------
__global__ void k_logits(const float* __restrict__ Lp, const float* __restrict__ epsp,
                         const float* __restrict__ mu, float* __restrict__ hard) {
    int wave = (blockIdx.x*blockDim.x + threadIdx.x) >> 5;  // 16*13 waves exactly
    int lane = threadIdx.x & 31;
    int b  = wave / 13;
    int tm = wave % 13;
    int hi = lane >> 4, l15 = lane & 15;
    const float* rA = Lp + ((size_t)b * NPAD + tm*16 + l15) * NSEG;
    const float* eb = epsp + (size_t)b * NSEG * MC;
    v8f c = {};
    #pragma unroll
    for (int ks = 0; ks < NSEG/4; ++ks) {
        int k = ks*4 + 2*hi;
        v2f a, bm;
        a.x  = rA[k];            a.y  = rA[k+1];
        bm.x = eb[k*MC + l15];   bm.y = eb[(k+1)*MC + l15];
        c = __builtin_amdgcn_wmma_f32_16x16x4_f32(false, a, false, bm,
                                                  (short)0, c, false, false);
    }
    #pragma unroll
    for (int r = 0; r < 8; ++r) {
        int s = tm*16 + r + 8*hi;
        int m = l15;
        if (s < NSEG) {
            float logit = c[r] + mu[b*NSEG + s];
            uint32_t h = hashu(((uint32_t)((b*NSEG + s)*MC + m)) * 2654435761u + 12345u);
            float u = fminf(fmaxf(u01(h), 1e-6f), 1.f - 1e-6f);
            float logistic = logf(u) - log1pf(-u);
            float relaxed  = sigmoidf(logit + logistic);     // TEMP = 1
            hard[((size_t)b*NSEG + s)*MC + m] = (relaxed > 0.5f) ? 1.f : 0.f;
        }
    }
}

// ---------------- K7: gather hard row per pixel, broadcast to (C, MC) ------
__global__ void k_mask(const int* __restrict__ groups, const float* __restrict__ hard,
                       float4* __restrict__ mask_out) {
    int idx = blockIdx.x*blockDim.x + threadIdx.x;   // B*HWP
    if (idx >= BB*HWP) return;
    int p = idx % HWP, b = idx / HWP;
    int g = groups[idx];
    const float4* hr = (const float4*)(hard + ((size_t)b*NSEG + g)*MC);
    float4 h0 = hr[0], h1 = hr[1], h2 = hr[2], h3 = hr[3];
    #pragma unroll
    for (int c = 0; c < CC; ++c) {
        float4* o = mask_out + (((size_t)(b*CC + c))*HWP + p)*4;
        o[0] = h0; o[1] = h1; o[2] = h2; o[3] = h3;
    }
}

extern "C" void kernel_launch(void* const* d_in, const int* in_sizes, int n_in,
                              void* d_out, int out_size, void* d_ws, size_t ws_size,
                              hipStream_t stream) {
    const float* x      = (const float*)d_in[0];
    const int*   groups = (const int*)  d_in[1];
    const float* Wc     = (const float*)d_in[2];
    const float* bc     = (const float*)d_in[3];

    // d_out layout (return order, flat): mask | group_probs | pixel_probs | sigma
    float* out   = (float*)d_out;
    float* mask  = out;
    float* gp    = mask + (size_t)BB*CC*HWP*MC;      // 38,535,168
    float* pp    = gp   + (size_t)BB*NSEG;
    float* sigma = pp   + (size_t)BB*HWP;

    // workspace layout (all f32)
    char* ws = (char*)d_ws;
    size_t off = 0;
    float* sums  = (float*)(ws + off); off += (size_t)BB*NSEG*EMB*sizeof(float);
    float* cnts  = (float*)(ws + off); off += (size_t)BB*NSEG*sizeof(float);
    float* mu    = (float*)(ws + off); off += (size_t)BB*NSEG*sizeof(float);
    float* sigL  = (float*)(ws + off); off += (size_t)BB*NSEG*NSEG*sizeof(float);
    float* hard  = (float*)(ws + off); off += (size_t)BB*NSEG*MC*sizeof(float);
    float* gembp = (float*)(ws + off); off += (size_t)BB*NPAD*KG*sizeof(float);
    float* Lp    = (float*)(ws + off); off += (size_t)BB*NPAD*NSEG*sizeof(float);
    float* epsp  = (float*)(ws + off); off += (size_t)BB*NSEG*MC*sizeof(float);

    // K0: zero accumulators (sums+cnts are contiguous)
    {
        int n = BB*NSEG*EMB + BB*NSEG;
        k_zero<<<(n + 255)/256, 256, 0, stream>>>(sums, n);
    }
    // K1: conv + pixel_probs + segment sums
    k_conv<<<dim3(HWP/1024, BB), 256, 0, stream>>>(x, groups, Wc, bc, pp, sums, cnts);
    // K2: averages
    k_avg<<<(BB*NPAD + 255)/256, 256, 0, stream>>>(sums, cnts, mu, gp, gembp);
    // K3: sigma via fp32 WMMA (16*169 waves, 8 waves/block)
    k_sigma<<<(BB*169)/8, 256, 0, stream>>>(gembp, sigma, sigL);
    // K4: Cholesky per batch
    k_chol<<<BB, 256, 0, stream>>>(sigL, Lp);
    // K5: eps
    {
        int n = BB*NSEG*MC;
        k_eps<<<(n + 255)/256, 256, 0, stream>>>(epsp);
    }
    // K6: logits via fp32 WMMA (16*13 waves, 8 waves/block)
    k_logits<<<(BB*13 + 7)/8, 256, 0, stream>>>(Lp, epsp, mu, hard);
    // K7: mask streaming write
    {
        int n = BB*HWP;
        k_mask<<<(n + 255)/256, 256, 0, stream>>>(groups, hard, (float4*)mask);
    }
}